// GANLoss_71227737637217
// MI455X (gfx1250) — compile-verified
//
#include <hip/hip_runtime.h>

// ---------------------------------------------------------------------------
// GAN-style gather loss:  out = -mean_i( prob[i, targets[i]] * reward[i] )
//   N = 16384 rows, C = 32000 cols (derived from in_sizes at launch)
//
// Memory-latency-bound gather; WMMA f32 16x16x4 (exact fp32 FMA tree) is used
// as the wave-level reduction so the reduction is full precision and the
// CDNA5 matrix path is exercised.
// ---------------------------------------------------------------------------

typedef __attribute__((ext_vector_type(2))) float v2f;
typedef __attribute__((ext_vector_type(8))) float v8f;

// Exact fp32 wave-level sum using V_WMMA_F32_16X16X4_F32.
// A = ones(16x4), B packs the 32 lane values (b.x) + zeros (b.y) into 4x16.
// D[r][c] = sum_k B[k][c]  -> VGPR0 holds colsum[lane & 15] in every lane
// (rows M=0 and M=8 are identical).  Four xor-shuffles (masks 8..1) then give
// the full wave sum in every lane.  Deterministic tree order.
__device__ __forceinline__ float wave_sum_wmma(float acc) {
  v2f a; a.x = 1.0f; a.y = 1.0f;      // A-matrix: all ones
  v2f b; b.x = acc;  b.y = 0.0f;      // B-matrix: one value per lane
  v8f c = {};                         // C = 0
  // 8 args: (neg_a, A, neg_b, B, c_mod, C, reuse_a, reuse_b)
  c = __builtin_amdgcn_wmma_f32_16x16x4_f32(
      /*neg_a=*/false, a, /*neg_b=*/false, b,
      /*c_mod=*/(short)0, c, /*reuse_a=*/false, /*reuse_b=*/false);
  float v = c[0];                     // colsum[lane & 15]
  v += __shfl_xor(v, 8, 32);
  v += __shfl_xor(v, 4, 32);
  v += __shfl_xor(v, 2, 32);
  v += __shfl_xor(v, 1, 32);
  return v;                           // wave total (all lanes)
}

// Stage 1: grid-stride gather + per-wave partial sums.
__global__ void __launch_bounds__(256)
gan_gather_partial(const float* __restrict__ prob,
                   const long long* __restrict__ targets,
                   const float* __restrict__ reward,
                   float* __restrict__ partial,
                   int n, long long ccols) {
  const int tid    = blockIdx.x * blockDim.x + threadIdx.x;
  const int stride = gridDim.x * blockDim.x;

  float acc = 0.0f;
  for (int i = tid; i < n; i += stride) {
    const long long t = targets[i];
    // random 128B-line gather from the 2 GB prob matrix
    acc = __builtin_fmaf(prob[(long long)i * ccols + t], reward[i], acc);
  }

  // All 256 threads of the block are active here (loop reconverged) -> EXEC
  // is all ones, as required by WMMA.
  const float wsum = wave_sum_wmma(acc);
  if ((threadIdx.x & 31) == 0) {
    partial[tid >> 5] = wsum;         // one partial per wave
  }
}

// Stage 2: single wave reduces all wave partials and writes -mean.
__global__ void __launch_bounds__(32)
gan_final_reduce(const float* __restrict__ partial, int npart,
                 float* __restrict__ out, float inv_n) {
  const int lane = threadIdx.x;       // 32 threads = 1 wave, all active
  float acc = 0.0f;
  for (int j = lane; j < npart; j += 32) acc += partial[j];
  const float total = wave_sum_wmma(acc);
  if (lane == 0) out[0] = -total * inv_n;
}

extern "C" void kernel_launch(void* const* d_in, const int* in_sizes, int n_in,
                              void* d_out, int out_size, void* d_ws, size_t ws_size,
                              hipStream_t stream) {
  const float*     prob    = (const float*)d_in[0];
  const long long* targets = (const long long*)d_in[1];   // jnp.int64
  const float*     reward  = (const float*)d_in[2];
  float*           out     = (float*)d_out;
  float*           partial = (float*)d_ws;

  const int       n     = in_sizes[2];                    // N (reward count)
  const long long ccols = (long long)in_sizes[0] / (long long)n;  // C

  constexpr int BLOCKS  = 64;   // 64 * 256 = 16384 threads: 1 row/thread at N=16384
  constexpr int THREADS = 256;  // 8 wave32 per block
  const int npart = (BLOCKS * THREADS) / 32;              // 512 wave partials

  gan_gather_partial<<<BLOCKS, THREADS, 0, stream>>>(prob, targets, reward,
                                                     partial, n, ccols);
  gan_final_reduce<<<1, 32, 0, stream>>>(partial, npart, out,
                                         1.0f / (float)n);
}